// GLES_21955872817609
// MI455X (gfx1250) — compile-verified
//
#include <hip/hip_runtime.h>

// Problem constants (from the reference)
constexpr int    NN    = 8192;     // rows of x == matrix dim (power of two)
constexpr int    DD    = 256;      // feature dim (K of the GEMM)
constexpr float  INV2L = 0.5f;     // 1/(2*LMBDA), LMBDA=1
constexpr float  EPS   = 1e-8f;    // sparse_plus eps
constexpr double EDGES = 32768.0;  // n * (int(log10(n)) + 1) = 8192*4
constexpr int    NPART = 1024;     // blocks for standalone reductions
constexpr int    GPART = 8192;     // blocks of the GEMM grid (fused reduction)

typedef __attribute__((ext_vector_type(2))) float v2f;
typedef __attribute__((ext_vector_type(4))) float v4f;
typedef __attribute__((ext_vector_type(8))) float v8f;

// sp(z) = (z + sqrt(z^2+eps))/2 and d(sum sp)/dgamma terms via one v_rsq_f32:
// rs = 1/sqrt(w), sqrt(w) = w*rs, z/sqrt(w) = z*rs.
__device__ inline void sp_terms(float z, float& s0f, float& s1f) {
  const float w  = __builtin_fmaf(z, z, EPS);
  const float rs = __builtin_amdgcn_rsqf(w);
  s0f += 0.5f * (z + w * rs);
  s1f += 0.5f * (1.0f + z * rs) * (-INV2L);
}

// ---------------------------------------------------------------------------
// Kernel 1: row squared norms; also initializes gamma = min(theta) = 0 exactly
// ---------------------------------------------------------------------------
__global__ void __launch_bounds__(256)
rownorm_kernel(const float* __restrict__ x, float* __restrict__ sq,
               float* __restrict__ gammap) {
  __shared__ float red[256];
  const int row = blockIdx.x;
  const float v = x[(size_t)row * DD + threadIdx.x];
  red[threadIdx.x] = v * v;
  __syncthreads();
#pragma unroll
  for (int s = 128; s > 0; s >>= 1) {
    if ((int)threadIdx.x < s) red[threadIdx.x] += red[threadIdx.x + s];
    __syncthreads();
  }
  if (threadIdx.x == 0) {
    sq[row] = red[0];
    if (row == 0) *gammap = 0.0f;  // min(theta) == 0 (masked diagonal)
  }
}

// ---------------------------------------------------------------------------
// Kernel 2: theta = sqrt(max(sq_i + sq_j - 2*dot, 0)), zero diagonal, via
// V_WMMA_F32_16X16X4_F32; wave computes a 32x32 tile, block = 8 waves (64x128).
// Fused: Newton iteration-0 sums at gamma==0 (theta already in registers).
// ---------------------------------------------------------------------------
__device__ inline void store_tile_acc(float* __restrict__ out,
                                      const float* __restrict__ sq,
                                      v8f c, int Ibase, int Jbase, int mi, int nj,
                                      double& s0, double& s1) {
  const int   i0  = Ibase + mi;
  const int   j   = Jbase + nj;
  const float sqj = sq[j];
  const float* sqi = sq + i0;
  float* p = out + (size_t)i0 * NN + j;   // rows are 32KB apart: imm offsets
  float s0f = 0.0f, s1f = 0.0f;
#pragma unroll
  for (int v = 0; v < 8; ++v) {
    float d2 = sqi[v] + sqj - 2.0f * c[v];
    d2 = fmaxf(d2, 0.0f);
    float th = __builtin_amdgcn_sqrtf(d2);  // raw v_sqrt_f32 (<=1 ulp)
    const bool diag = (i0 + v == j);
    if (diag) th = 0.0f;
    p[(size_t)v * NN] = th;
    if (diag) {
      s0f += 0.5f * sqrtf(EPS);             // sp(0); zero gradient (const-folded)
    } else {
      sp_terms(-th * INV2L, s0f, s1f);      // gamma == 0 for iteration 0
    }
  }
  s0 += (double)s0f;
  s1 += (double)s1f;
}

__global__ void __launch_bounds__(256)
gemm_theta_kernel(const float* __restrict__ x, const float* __restrict__ sq,
                  float* __restrict__ out,
                  double* __restrict__ P0, double* __restrict__ P1) {
  const int lane = threadIdx.x & 31;
  const int wave = threadIdx.x >> 5;   // 0..7
  const int wrow = wave >> 2;          // 0..1
  const int wcol = wave & 3;           // 0..3
  const int I0 = blockIdx.y * 64  + wrow * 32;
  const int J0 = blockIdx.x * 128 + wcol * 32;

  const int m  = lane & 15;            // matrix row within 16
  const int kk = (lane >> 4) << 1;     // K offset 0 or 2 (hi half-wave)

  const float* a0p = x + (size_t)(I0 +      m) * DD + kk;
  const float* a1p = x + (size_t)(I0 + 16 + m) * DD + kk;
  const float* b0p = x + (size_t)(J0 +      m) * DD + kk;
  const float* b1p = x + (size_t)(J0 + 16 + m) * DD + kk;

  v8f c00 = {}, c01 = {}, c10 = {}, c11 = {};

#pragma unroll 4
  for (int k = 0; k < DD; k += 4) {
    const v2f a0 = *(const v2f*)(a0p + k);
    const v2f a1 = *(const v2f*)(a1p + k);
    const v2f b0 = *(const v2f*)(b0p + k);
    const v2f b1 = *(const v2f*)(b1p + k);
    c00 = __builtin_amdgcn_wmma_f32_16x16x4_f32(false, a0, false, b0, (short)0, c00, false, false);
    c01 = __builtin_amdgcn_wmma_f32_16x16x4_f32(false, a0, false, b1, (short)0, c01, false, false);
    c10 = __builtin_amdgcn_wmma_f32_16x16x4_f32(false, a1, false, b0, (short)0, c10, false, false);
    c11 = __builtin_amdgcn_wmma_f32_16x16x4_f32(false, a1, false, b1, (short)0, c11, false, false);
  }

  const int nj = lane & 15;
  const int mi = (lane >> 4) << 3;     // hi lanes -> M += 8
  double s0 = 0.0, s1 = 0.0;
  store_tile_acc(out, sq, c00, I0,      J0,      mi, nj, s0, s1);
  store_tile_acc(out, sq, c01, I0,      J0 + 16, mi, nj, s0, s1);
  store_tile_acc(out, sq, c10, I0 + 16, J0,      mi, nj, s0, s1);
  store_tile_acc(out, sq, c11, I0 + 16, J0 + 16, mi, nj, s0, s1);

  // Deterministic block reduction of the iteration-0 Newton sums
  __shared__ double r0[256];
  __shared__ double r1[256];
  r0[threadIdx.x] = s0;
  r1[threadIdx.x] = s1;
  __syncthreads();
#pragma unroll
  for (int s = 128; s > 0; s >>= 1) {
    if ((int)threadIdx.x < s) {
      r0[threadIdx.x] += r0[threadIdx.x + s];
      r1[threadIdx.x] += r1[threadIdx.x + s];
    }
    __syncthreads();
  }
  if (threadIdx.x == 0) {
    const int bid = blockIdx.y * gridDim.x + blockIdx.x;
    P0[bid] = r0[0];
    P1[bid] = r1[0];
  }
}

// ---------------------------------------------------------------------------
// Kernel 3: standalone Newton reduction (iterations 1..3), float4 streaming.
// ---------------------------------------------------------------------------
__global__ void __launch_bounds__(256)
newton_reduce_kernel(const float* __restrict__ theta,
                     const float* __restrict__ gammap,
                     double* __restrict__ P0, double* __restrict__ P1) {
  __shared__ double r0[256];
  __shared__ double r1[256];
  const float gamma = *gammap;
  const size_t total4 = (size_t)NN * NN / 4;
  const size_t stride = (size_t)gridDim.x * blockDim.x;
  double s0 = 0.0, s1 = 0.0;
  for (size_t q = (size_t)blockIdx.x * blockDim.x + threadIdx.x;
       q < total4; q += stride) {
    const size_t idx = q << 2;
    const int i     = (int)(idx >> 13);        // NN = 2^13
    const int jbase = (int)(idx & (NN - 1));   // quartet never crosses a row
    const v4f t = ((const v4f*)theta)[q];
    float s0f = 0.0f, s1f = 0.0f;
#pragma unroll
    for (int e = 0; e < 4; ++e) {
      if (i == jbase + e) {
        s0f += 0.5f * sqrtf(EPS);
      } else {
        sp_terms(-(t[e] + gamma) * INV2L, s0f, s1f);
      }
    }
    s0 += (double)s0f;
    s1 += (double)s1f;
  }
  r0[threadIdx.x] = s0;
  r1[threadIdx.x] = s1;
  __syncthreads();
#pragma unroll
  for (int s = 128; s > 0; s >>= 1) {
    if ((int)threadIdx.x < s) {
      r0[threadIdx.x] += r0[threadIdx.x + s];
      r1[threadIdx.x] += r1[threadIdx.x + s];
    }
    __syncthreads();
  }
  if (threadIdx.x == 0) {
    P0[blockIdx.x] = r0[0];
    P1[blockIdx.x] = r1[0];
  }
}

// ---------------------------------------------------------------------------
// Kernel 4: deterministic combine (strided serial + tree) + Newton update.
// ---------------------------------------------------------------------------
__global__ void __launch_bounds__(256)
newton_update_kernel(const double* __restrict__ P0, const double* __restrict__ P1,
                     int nparts, float* __restrict__ gammap) {
  __shared__ double r0[256];
  __shared__ double r1[256];
  double s0 = 0.0, s1 = 0.0;
  for (int i = threadIdx.x; i < nparts; i += 256) { s0 += P0[i]; s1 += P1[i]; }
  r0[threadIdx.x] = s0;
  r1[threadIdx.x] = s1;
  __syncthreads();
#pragma unroll
  for (int s = 128; s > 0; s >>= 1) {
    if ((int)threadIdx.x < s) {
      r0[threadIdx.x] += r0[threadIdx.x + s];
      r1[threadIdx.x] += r1[threadIdx.x + s];
    }
    __syncthreads();
  }
  if (threadIdx.x == 0) {
    const double equ = r0[0] - EDGES;
    *gammap = (float)((double)(*gammap) - equ / r1[0]);
  }
}

// ---------------------------------------------------------------------------
// Kernel 5: in-place finalize: out = relu(-(theta+gamma)/2L), zero diagonal.
// theta symmetric -> (G+G^T)/2 == G. float4 streaming.
// ---------------------------------------------------------------------------
__global__ void __launch_bounds__(256)
finalize_kernel(float* __restrict__ out, const float* __restrict__ gammap) {
  const float gamma = *gammap;
  const size_t q   = (size_t)blockIdx.x * blockDim.x + threadIdx.x;
  const size_t idx = q << 2;
  const int i     = (int)(idx >> 13);
  const int jbase = (int)(idx & (NN - 1));
  v4f t = ((const v4f*)out)[q];
#pragma unroll
  for (int e = 0; e < 4; ++e) {
    t[e] = (i == jbase + e) ? 0.0f : fmaxf(-(t[e] + gamma) * INV2L, 0.0f);
  }
  ((v4f*)out)[q] = t;
}

// ---------------------------------------------------------------------------
extern "C" void kernel_launch(void* const* d_in, const int* in_sizes, int n_in,
                              void* d_out, int out_size, void* d_ws, size_t ws_size,
                              hipStream_t stream) {
  (void)in_sizes; (void)n_in; (void)out_size; (void)ws_size;

  const float* x = (const float*)d_in[0];
  float* out = (float*)d_out;

  // Workspace: [0,32KB) sq[8192] f32 | gamma f32 | pad | fp64 partials (2x8192)
  float*  sq     = (float*)d_ws;
  float*  gammap = sq + NN;
  double* P0     = (double*)((char*)d_ws + 33280);  // 8B aligned
  double* P1     = P0 + GPART;

  // 1) row norms + gamma init
  rownorm_kernel<<<NN, 256, 0, stream>>>(x, sq, gammap);

  // 2) theta via fp32 WMMA into d_out, with fused iteration-0 Newton sums
  dim3 ggrid(NN / 128, NN / 64);   // 64 x 128 = 8192 blocks
  gemm_theta_kernel<<<ggrid, 256, 0, stream>>>(x, sq, out, P0, P1);
  newton_update_kernel<<<1, 256, 0, stream>>>(P0, P1, GPART, gammap);

  // 3) Newton iterations 1..3 (5th reference iteration doesn't affect output)
  for (int it = 0; it < 3; ++it) {
    newton_reduce_kernel<<<NPART, 256, 0, stream>>>(out, gammap, P0, P1);
    newton_update_kernel<<<1, 256, 0, stream>>>(P0, P1, NPART, gammap);
  }

  // 4) in-place final transform (float4 lanes)
  finalize_kernel<<<(unsigned)((size_t)NN * NN / 4 / 256), 256, 0, stream>>>(out, gammap);
}